// ChamLoss_32195074851325
// MI455X (gfx1250) — compile-verified
//
#include <hip/hip_runtime.h>
#include <math.h>

// CDNA5 wave32 WMMA fragment types
typedef __attribute__((ext_vector_type(2))) float v2f;
typedef __attribute__((ext_vector_type(8))) float v8f;

#define WAVES   4
#define THREADS (WAVES * 32)
#define CHUNK   1024   // y points staged per iteration: 2 x float2 arrays = 16 KB

__global__ void zero_out_kernel(float* out) { out[0] = 0.0f; }

// For each x point: min over all y of squared distance, then sqrt/sum/scale ->
// atomicAdd into out[0].
//
// WMMA mapping (V_WMMA_F32_16X16X4_F32, D = A*B + C, C = 0):
//   A(m,:) = (x0, x1, x2, 1)
//   B(:,n) = (-2y0, -2y1, -2y2, ||y||^2)
//   => D(m,n) = ||y_n||^2 - 2 x_m.y_n
// min_n d(m,n) = ||x_m||^2 + min_n D(m,n)   (||x||^2 added AFTER the reduction)
//
// Fragment layout (CDNA5 ISA, 32-bit 16x4): lane&15 = M (A) / N (B);
// lanes 0-15 hold K={0,1}, lanes 16-31 hold K={2,3}.
// B fragments are pre-baked in LDS so the inner loop B setup is one ds_load_b64.
__global__ __launch_bounds__(THREADS)
void chamfer_min_wmma(const float* __restrict__ x, const float* __restrict__ y,
                      int Nx, int Ny, int blocks_per_batch,
                      float scale_base, const float* __restrict__ alpha,
                      float* __restrict__ out)
{
    __shared__ float2 sb[2 * CHUNK];   // [0,CHUNK): (-2y0,-2y1)  [CHUNK,2C): (-2y2,||y||^2)
    __shared__ float  red[WAVES * 16]; // per-wave per-m min staging

    const int tid  = threadIdx.x;
    const int wave = tid >> 5;
    const int lane = tid & 31;
    const int l15  = lane & 15;
    const bool hi  = lane >= 16;

    const int batch  = blockIdx.x / blocks_per_batch;
    const int blk_in = blockIdx.x % blocks_per_batch;
    const int x_base = (blk_in * WAVES + wave) * 16;

    // ---- A fragment: this wave's 16 x-points (homogeneous row, norm deferred)
    const float* xp = x + (size_t)(batch * Nx + x_base + l15) * 3;
    const float x0 = xp[0], x1 = xp[1], x2 = xp[2];
    const float xn = x0 * x0 + x1 * x1 + x2 * x2;
    v2f a;
    a[0] = hi ? x2   : x0;   // K2 : K0
    a[1] = hi ? 1.0f : x1;   // K3 : K1

    v8f acc;
#pragma unroll
    for (int k = 0; k < 8; ++k) acc[k] = 3.0e38f;
    const v8f czero = {};    // C = 0, loop-invariant

    const float* yb = y + (size_t)batch * Ny * 3;
    const float2* bp = sb + (hi ? CHUNK : 0) + l15;  // this lane's fragment base

    for (int base = 0; base < Ny; base += CHUNK) {
        int clen = Ny - base;
        if (clen > CHUNK) clen = CHUNK;

        __syncthreads();
        // Stage pre-baked B fragments: all -2y / norm math done once per chunk
        for (int i = tid; i < clen; i += THREADS) {
            const float* yp = yb + (size_t)(base + i) * 3;
            const float w0 = yp[0], w1 = yp[1], w2 = yp[2];
            sb[i]         = make_float2(-2.0f * w0, -2.0f * w1);
            sb[CHUNK + i] = make_float2(-2.0f * w2, w0 * w0 + w1 * w1 + w2 * w2);
        }
        __syncthreads();

        if (clen == CHUNK) {           // hot path: compile-time trip count
#pragma unroll 8
            for (int t = 0; t < CHUNK / 16; ++t) {
                const v2f b = *(const v2f*)(bp + t * 16);
                const v8f d = __builtin_amdgcn_wmma_f32_16x16x4_f32(
                    false, a, false, b, (short)0, czero, false, false);
#pragma unroll
                for (int k = 0; k < 8; ++k) acc[k] = fminf(acc[k], d[k]);
            }
        } else {
            const int ntiles = clen >> 4;
            for (int t = 0; t < ntiles; ++t) {
                const v2f b = *(const v2f*)(bp + t * 16);
                const v8f d = __builtin_amdgcn_wmma_f32_16x16x4_f32(
                    false, a, false, b, (short)0, czero, false, false);
#pragma unroll
                for (int k = 0; k < 8; ++k) acc[k] = fminf(acc[k], d[k]);
            }
        }
    }

    // min across the 16 lanes of each half-wave (the n dimension)
#pragma unroll
    for (int off = 1; off < 16; off <<= 1) {
#pragma unroll
        for (int k = 0; k < 8; ++k)
            acc[k] = fminf(acc[k], __shfl_xor(acc[k], off, 32));
    }

    // Put per-m mins back in lane order: red[wave*16 + m], m=0..7 from lower
    // half (M=k), m=8..15 from upper half (M=k+8).
    if (l15 == 0) {
        float* r = &red[wave * 16 + (hi ? 8 : 0)];
#pragma unroll
        for (int k = 0; k < 8; ++k) r[k] = acc[k];
    }
    __syncthreads();

    // lane l15 owns x point m=l15: add its ||x||^2, clamp, sqrt
    float dval = sqrtf(fmaxf(red[wave * 16 + l15] + xn, 0.0f));
    // sum over the 16 m's (upper half holds duplicates; only lane 0 commits)
#pragma unroll
    for (int off = 1; off < 16; off <<= 1)
        dval += __shfl_xor(dval, off, 32);

    if (lane == 0) {
        const float scale = scale_base * (alpha ? alpha[0] : 1.0f);
        atomicAdd(out, dval * scale);
    }
}

extern "C" void kernel_launch(void* const* d_in, const int* in_sizes, int n_in,
                              void* d_out, int out_size, void* d_ws, size_t ws_size,
                              hipStream_t stream)
{
    const float* coarse = (const float*)d_in[0];
    const float* fine   = (const float*)d_in[1];
    const float* gt     = (const float*)d_in[2];
    const float* alpha  = (const float*)d_in[3];
    float* out = (float*)d_out;

    const int B  = 4;
    const int Nc = in_sizes[0] / (B * 3);   // 1024
    const int Nf = in_sizes[1] / (B * 3);   // 8192
    const int Ng = in_sizes[2] / (B * 3);   // 8192

    zero_out_kernel<<<1, 1, 0, stream>>>(out);

    auto launch = [&](const float* x, const float* y, int Nx, int Ny,
                      float w, const float* al) {
        const int bpb = Nx / (16 * WAVES);
        chamfer_min_wmma<<<dim3(B * bpb), dim3(THREADS), 0, stream>>>(
            x, y, Nx, Ny, bpb, w / (float)((size_t)B * Nx), al, out);
    };

    // _dist_cd(coarse, gt) = mean_gt sqrt(min over coarse) + 0.1*mean_coarse sqrt(min over gt)
    launch(gt,     coarse, Ng, Nc, 1.0f, nullptr);
    launch(coarse, gt,     Nc, Ng, 0.1f, nullptr);
    // alpha * _dist_cd(fine, gt)
    launch(gt,     fine,   Ng, Nf, 1.0f, alpha);
    launch(fine,   gt,     Nf, Ng, 0.1f, alpha);
}